// Method5_Adaptive_54400055771990
// MI455X (gfx1250) — compile-verified
//
#include <hip/hip_runtime.h>
#include <stdint.h>

// ---------------------------------------------------------------------------
// softplus + per-sample max-normalization, fp32, shape [128, 1, 512, 512].
// Bandwidth-bound: >=268MB HBM traffic vs ~0.5 GFLOP. Strategy:
//   pass 1: per-sample max of RAW x (softplus is monotone) via async
//           global->LDS b128 copies (ASYNCcnt pipeline, CDNA5 path)
//   pass 2: tiny kernel: softplus(max), reciprocal scale w/ EPS guard
//   pass 3: softplus(x) * scale, b128 loads (hit 192MB L2) + NT b128 stores
// ---------------------------------------------------------------------------

typedef __attribute__((ext_vector_type(4))) float v4f;

#define NSAMPLES    128
#define RED_THREADS 256
#define RED_K       16          // float4 async loads per thread (16-deep pipeline)
#define APP_THREADS 256
#define APP_ITEMS   4
#define EPSF        1e-8f

__device__ __forceinline__ float softplus_f32(float x) {
    // numerically stable: max(x,0) + log1p(exp(-|x|))  (matches jax.nn.softplus)
    return fmaxf(x, 0.0f) + log1pf(expf(-fabsf(x)));
}

// ---------------------------------------------------------------------------
// Pass 1: per-(sample, chunk) max of raw x.
// grid = (chunks_per_sample, NSAMPLES), block = 256.
// Each block streams RED_K*256 float4 (64KB) through LDS with async copies.
// ---------------------------------------------------------------------------
__global__ void __launch_bounds__(RED_THREADS)
reduce_max_kernel(const v4f* __restrict__ x4, float* __restrict__ ws_max) {
    __shared__ v4f lbuf[RED_K][RED_THREADS];      // 64 KB staging (WGP has 320 KB)
    const int tid = threadIdx.x;
    const size_t base =
        ((size_t)blockIdx.y * (size_t)gridDim.x + (size_t)blockIdx.x) *
            (size_t)(RED_K * RED_THREADS) + (size_t)tid;

    // Issue the full 16-deep async global->LDS pipeline (tracked by ASYNCcnt;
    // completions return in order, so counted waits below are exact).
    #pragma unroll
    for (int k = 0; k < RED_K; ++k) {
        uint32_t lds_off = (uint32_t)(uintptr_t)&lbuf[k][tid]; // low 32b of flat = LDS addr
        const v4f* g = x4 + base + (size_t)k * RED_THREADS;
        asm volatile("global_load_async_to_lds_b128 %0, %1, off"
                     :: "v"(lds_off), "v"(g) : "memory");
    }

    float m = -__builtin_inff();
    #pragma unroll
    for (int k = 0; k < RED_K; ++k) {
        asm volatile("s_wait_asynccnt %0" :: "i"(RED_K - 1 - k) : "memory");
        v4f v = lbuf[k][tid];
        m = fmaxf(fmaxf(fmaxf(v.x, v.y), fmaxf(v.z, v.w)), m);
    }

    // wave32 butterfly reduction
    #pragma unroll
    for (int off = 16; off > 0; off >>= 1)
        m = fmaxf(m, __shfl_xor(m, off, 32));

    __shared__ float wred[RED_THREADS / 32];
    if ((tid & 31) == 0) wred[tid >> 5] = m;
    __syncthreads();
    if (tid == 0) {
        float r = wred[0];
        #pragma unroll
        for (int w = 1; w < RED_THREADS / 32; ++w) r = fmaxf(r, wred[w]);
        ws_max[blockIdx.y * gridDim.x + blockIdx.x] = r;
    }
}

// ---------------------------------------------------------------------------
// Pass 2: fold chunk maxes, compute per-sample scale. 1 block, 128 threads.
// ---------------------------------------------------------------------------
__global__ void scale_kernel(const float* __restrict__ ws_max,
                             float* __restrict__ ws_scale, int cpb) {
    const int s = threadIdx.x;
    float m = ws_max[s * cpb];
    for (int i = 1; i < cpb; ++i) m = fmaxf(m, ws_max[s * cpb + i]);
    const float sp = softplus_f32(m);       // == max over softplus (monotone)
    ws_scale[s] = (sp > EPSF) ? (1.0f / sp) : 1.0f;
}

// ---------------------------------------------------------------------------
// Pass 3: out = softplus(x) * scale[sample].
// grid = (per_sample4/(256*4), NSAMPLES). x re-read hits L2 (tensor fits in
// 192MB); output uses non-temporal b128 stores to avoid evicting x lines.
// ---------------------------------------------------------------------------
__global__ void __launch_bounds__(APP_THREADS)
apply_kernel(const v4f* __restrict__ x4, const float* __restrict__ ws_scale,
             v4f* __restrict__ out4) {
    const int tid = threadIdx.x;
    const float s = ws_scale[blockIdx.y];   // uniform per block -> scalar load
    const size_t base =
        ((size_t)blockIdx.y * (size_t)gridDim.x + (size_t)blockIdx.x) *
            (size_t)(APP_THREADS * APP_ITEMS) + (size_t)tid;

    v4f v[APP_ITEMS];
    #pragma unroll
    for (int i = 0; i < APP_ITEMS; ++i)
        v[i] = x4[base + i * APP_THREADS];

    #pragma unroll
    for (int i = 0; i < APP_ITEMS; ++i) {
        v4f r;
        r.x = softplus_f32(v[i].x) * s;
        r.y = softplus_f32(v[i].y) * s;
        r.z = softplus_f32(v[i].z) * s;
        r.w = softplus_f32(v[i].w) * s;
        __builtin_nontemporal_store(r, out4 + base + i * APP_THREADS);
    }
}

// ---------------------------------------------------------------------------
extern "C" void kernel_launch(void* const* d_in, const int* in_sizes, int n_in,
                              void* d_out, int out_size, void* d_ws, size_t ws_size,
                              hipStream_t stream) {
    const float* x = (const float*)d_in[0];
    float* out = (float*)d_out;

    const int total       = in_sizes[0];            // 128*512*512 = 33554432
    const int per_sample  = total / NSAMPLES;       // 262144
    const int per_sample4 = per_sample / 4;         // 65536

    float* ws_max   = (float*)d_ws;                 // NSAMPLES * cpb floats
    float* ws_scale = ws_max + NSAMPLES * (per_sample4 / (RED_K * RED_THREADS));

    dim3 gRed(per_sample4 / (RED_K * RED_THREADS), NSAMPLES);     // (16, 128)
    reduce_max_kernel<<<gRed, RED_THREADS, 0, stream>>>((const v4f*)x, ws_max);

    scale_kernel<<<1, NSAMPLES, 0, stream>>>(ws_max, ws_scale, (int)gRed.x);

    dim3 gApp(per_sample4 / (APP_THREADS * APP_ITEMS), NSAMPLES); // (64, 128)
    apply_kernel<<<gApp, APP_THREADS, 0, stream>>>((const v4f*)x, ws_scale, (v4f*)out);
}